// SpatioTemporal_map_80951543595326
// MI455X (gfx1250) — compile-verified
//
#include <hip/hip_runtime.h>

#define B_  4
#define T_  8
#define C_  64
#define H_  32
#define W_  32
#define F_  16
#define HW_ 1024        // H*W
#define N_  8192        // T*H*W

typedef float v2f __attribute__((ext_vector_type(2)));
typedef float v8f __attribute__((ext_vector_type(8)));

// ---------------------------------------------------------------------------
// Kernel 1: pooled[b,n] = mean_f( conv1(x) )  ==  wbar . x[b,:,n] + bbar
// Output padded to 16 "batch" rows (rows 4..15 = 0) so the WMMA A operand in
// kernel 2 needs no lane masking.
// ---------------------------------------------------------------------------
__global__ void __launch_bounds__(256)
pool_kernel(const float* __restrict__ x,
            const float* __restrict__ conv1_w,   // [F, C]
            const float* __restrict__ conv1_b,   // [F]
            float* __restrict__ pooledP)         // [16, N]
{
    __shared__ float wbar[C_];
    __shared__ float bbar;
    const int tid = threadIdx.x;
    if (tid < C_) {
        float s = 0.f;
        #pragma unroll
        for (int f = 0; f < F_; ++f) s += conv1_w[f * C_ + tid];
        wbar[tid] = s * (1.0f / F_);
    }
    if (tid == 0) {
        float s = 0.f;
        #pragma unroll
        for (int f = 0; f < F_; ++f) s += conv1_b[f];
        bbar = s * (1.0f / F_);
    }
    __syncthreads();

    const int idx = blockIdx.x * 256 + tid;      // 0 .. 16*N-1
    const int bb  = idx >> 13;                   // padded batch row
    const int n   = idx & (N_ - 1);
    if (bb >= B_) { pooledP[idx] = 0.f; return; }

    const int t  = n / HW_;
    const int hw = n % HW_;
    const float* xp = x + (size_t)((bb * T_ + t) * C_) * HW_ + hw;

    float acc = bbar;
    #pragma unroll
    for (int c = 0; c < C_; ++c) acc += wbar[c] * xp[c * HW_];
    pooledP[idx] = acc;
}

// ---------------------------------------------------------------------------
// Kernel 2: v[b,m] = sigmoid( sum_k pooled[b,k] * Wt[m,k] + bias[m] )
// HBM-bound on streaming Wt (268 MB > 192 MB L2, hence NT loads for W so the
// one-shot stream does not evict pooledP/x). One 16-row m-tile per block;
// 8 waves split K; per iteration: 1x b64 load of pooled (A, L2-resident),
// 1x NT b64 load of W (B), 1x v_wmma_f32_16x16x4_f32.
// D[b=0..3][m0+j] lands in acc[b] of lane j (rows 4..15 zero by padding).
// ---------------------------------------------------------------------------
__global__ void __launch_bounds__(256)
gemv_wmma_kernel(const float* __restrict__ pooledP,  // [16, N]
                 const float* __restrict__ Wt,       // ffnn1_w [N, N]
                 const float* __restrict__ bias,     // [N]
                 float* __restrict__ vbuf)           // [4, N]
{
    __shared__ float red[8 * 64];                    // [wave][b][j]
    const int tid  = threadIdx.x;
    const int wave = tid >> 5;
    const int lane = tid & 31;
    const int r    = lane & 15;                      // row within 16
    const int half = lane >> 4;                      // K sub-pair select
    const int m0   = blockIdx.x * 16;

    const int k0 = wave * (N_ / 8);                  // 1024-wide K slice
    const float* aptr = pooledP + r * N_ + k0 + 2 * half;
    const float* bptr = Wt + (size_t)(m0 + r) * N_ + k0 + 2 * half;

    v8f acc = {};
    #pragma unroll 8
    for (int kk = 0; kk < N_ / 8; kk += 4) {         // 256 iterations, K step 4
        v2f a = *(const v2f*)aptr;                            // pooled (RT: cache)
        v2f b = __builtin_nontemporal_load((const v2f*)bptr); // W stream (NT)
        acc = __builtin_amdgcn_wmma_f32_16x16x4_f32(
                  false, a, false, b, (short)0, acc, false, false);
        aptr += 4; bptr += 4;
    }

    // lanes 0..15 hold D[b][m0+lane] in acc[b] (b = VGPR index 0..3 valid)
    if (lane < 16) {
        red[wave * 64 + 0 * 16 + lane] = acc[0];
        red[wave * 64 + 1 * 16 + lane] = acc[1];
        red[wave * 64 + 2 * 16 + lane] = acc[2];
        red[wave * 64 + 3 * 16 + lane] = acc[3];
    }
    __syncthreads();

    if (tid < 64) {
        const int j = tid & 15;
        const int b = tid >> 4;
        float s = 0.f;
        #pragma unroll
        for (int w = 0; w < 8; ++w) s += red[w * 64 + b * 16 + j];
        s += bias[m0 + j];
        vbuf[b * N_ + m0 + j] = 1.0f / (1.0f + __expf(-s));
    }
}

// ---------------------------------------------------------------------------
// Kernel 3: out[b,t,f,h,w] = v[b,n] * ( g3_w . x[b,t,:,h,w] + g3_b[f] )
// Output is write-once -> non-temporal stores.
// ---------------------------------------------------------------------------
__global__ void __launch_bounds__(256)
out_kernel(const float* __restrict__ x,
           const float* __restrict__ g3_w,    // [F, C]
           const float* __restrict__ g3_b,    // [F]
           const float* __restrict__ vbuf,    // [4, N]
           float* __restrict__ out)           // [B, T, F, H, W]
{
    __shared__ float gw[F_ * C_];
    __shared__ float gb[F_];
    const int tid = threadIdx.x;
    for (int i = tid; i < F_ * C_; i += 256) gw[i] = g3_w[i];
    if (tid < F_) gb[tid] = g3_b[tid];
    __syncthreads();

    const int idx = blockIdx.x * 256 + tid;   // 0 .. B*N-1
    const int b   = idx >> 13;
    const int n   = idx & (N_ - 1);
    const int t   = n / HW_;
    const int hw  = n % HW_;
    const float* xp = x + (size_t)((b * T_ + t) * C_) * HW_ + hw;

    float acc[F_];
    #pragma unroll
    for (int f = 0; f < F_; ++f) acc[f] = gb[f];
    for (int c = 0; c < C_; ++c) {
        const float xv = xp[c * HW_];
        #pragma unroll
        for (int f = 0; f < F_; ++f) acc[f] += gw[f * C_ + c] * xv;
    }

    const float v = vbuf[b * N_ + n];
    float* op = out + (size_t)((b * T_ + t) * F_) * HW_ + hw;
    #pragma unroll
    for (int f = 0; f < F_; ++f)
        __builtin_nontemporal_store(v * acc[f], op + f * HW_);
}

// ---------------------------------------------------------------------------
extern "C" void kernel_launch(void* const* d_in, const int* in_sizes, int n_in,
                              void* d_out, int out_size, void* d_ws, size_t ws_size,
                              hipStream_t stream) {
    const float* x       = (const float*)d_in[0];
    // d_in[1] (x1) unused by reference
    const float* conv1_w = (const float*)d_in[2];
    const float* conv1_b = (const float*)d_in[3];
    const float* g3_w    = (const float*)d_in[4];
    const float* g3_b    = (const float*)d_in[5];
    const float* ffnn1_w = (const float*)d_in[6];
    const float* ffnn1_b = (const float*)d_in[7];

    float* pooledP = (float*)d_ws;            // 16*N floats  = 512 KB
    float* vbuf    = pooledP + 16 * N_;       //  4*N floats  = 128 KB

    pool_kernel<<<(16 * N_) / 256, 256, 0, stream>>>(x, conv1_w, conv1_b, pooledP);
    gemv_wmma_kernel<<<N_ / 16, 256, 0, stream>>>(pooledP, ffnn1_w, ffnn1_b, vbuf);
    out_kernel<<<(B_ * N_) / 256, 256, 0, stream>>>(x, g3_w, g3_b, vbuf, (float*)d_out);
}